// QNN_19207093747726
// MI455X (gfx1250) — compile-verified
//
#include <hip/hip_runtime.h>

// ---------------------------------------------------------------------------
// 16-qubit QNN statevector simulator for MI455X (gfx1250, wave32, WMMA).
//
// One workgroup (256 threads = 8 wave32) per circuit replica m = o*16 + b.
// Full 2^16-amplitude f32 statevector lives in LDS (256 KB <= 320 KB/WGP).
// Four adjacent RY gates are fused into a 16x16 unitary (Kronecker product),
// applied to the state via chained V_WMMA_F32_16X16X4_F32 (K = 4*4 = 16).
// ---------------------------------------------------------------------------

typedef float v2f __attribute__((ext_vector_type(2)));
typedef float v8f __attribute__((ext_vector_type(8)));

#define NQ       16
#define DIM      65536          // 2^16 amplitudes
#define NLAYERS  4              // variational layers
#define OUTF     8
#define NB       16
#define THREADS  256            // 8 wave32 waves
#define NWAVES   8

__global__ __launch_bounds__(THREADS)
void qnn_statevector_wmma(const float* __restrict__ x,     // [NB, NQ]
                          const float* __restrict__ w,     // [OUTF, NQ, NLAYERS]
                          float* __restrict__ out)         // [OUTF, NB] flat
{
    extern __shared__ float smem[];
    float* psi = smem;            // DIM floats: the statevector
    float* U   = smem + DIM;      // 256 floats: fused 16x16 group unitary
    float* red = smem + DIM;      // reused for final reduction

    const int m    = blockIdx.x;      // replica id = o*NB + b
    const int o    = m / NB;
    const int b    = m % NB;
    const int tid  = threadIdx.x;
    const int lane = tid & 31;
    const int wave = tid >> 5;
    const int half = lane >> 4;       // 0: lanes 0-15, 1: lanes 16-31
    const int lcol = lane & 15;

    // ---- init |0...0> ----
    for (int i = tid; i < DIM; i += THREADS) psi[i] = 0.0f;
    __syncthreads();
    if (tid == 0) psi[0] = 1.0f;
    __syncthreads();

    // ---- 5 gate layers: layer 0 = data encoding RY(x), layers 1..4 = RY(w) ----
    for (int layer = 0; layer < 1 + NLAYERS; ++layer) {
        for (int g = 0; g < 4; ++g) {
            // Build U_g[r][c] = prod_j R(theta_{4g+j})[bit_{3-j}(r)][bit_{3-j}(c)]
            // R(t) = [[cos(t/2), -sin(t/2)], [sin(t/2), cos(t/2)]]
            {
                const int r = tid >> 4;
                const int c = tid & 15;
                float val = 1.0f;
#pragma unroll
                for (int j = 0; j < 4; ++j) {
                    const int q = 4 * g + j;
                    const float th = (layer == 0) ? x[b * NQ + q]
                                                  : w[(o * NQ + q) * NLAYERS + (layer - 1)];
                    const float ch = __cosf(0.5f * th);
                    const float sh = __sinf(0.5f * th);
                    const int rb = (r >> (3 - j)) & 1;
                    const int cb = (c >> (3 - j)) & 1;
                    val *= (rb == cb) ? ch : (rb ? sh : -sh);
                }
                U[tid] = val;
            }
            __syncthreads();

            // A fragments: 16x4 f32 chunks of U. Lanes 0-15 hold rows M with
            // K = {4kc, 4kc+1}; lanes 16-31 hold K = {4kc+2, 4kc+3}.
            v2f A[4];
#pragma unroll
            for (int kc = 0; kc < 4; ++kc) {
                const int k0 = 4 * kc + 2 * half;
                A[kc].x = U[lcol * 16 + k0];
                A[kc].y = U[lcol * 16 + k0 + 1];
            }

            // Strided view: idx = hi*(16*right) + m4*right + lo, right = 2^(12-4g)
            const int rightShift   = 12 - 4 * g;
            const unsigned right   = 1u << rightShift;
            const unsigned loMask  = right - 1u;

            // 4096 columns -> 256 tiles of 16 columns; waves stride by 8.
            for (int t = wave; t < 256; t += NWAVES) {
                const int      col  = t * 16 + lcol;
                const unsigned hi   = (unsigned)col >> rightShift;
                const unsigned lo   = (unsigned)col & loMask;
                const unsigned base = (hi << (rightShift + 4)) + lo;

                v8f acc = {0.f, 0.f, 0.f, 0.f, 0.f, 0.f, 0.f, 0.f};
#pragma unroll
                for (int kc = 0; kc < 4; ++kc) {
                    // B chunk: 4x16 rows of the psi tile (K rows = group value m4)
                    const int k0 = 4 * kc + 2 * half;
                    v2f B;
                    B.x = psi[base + (unsigned)(k0)     * right];
                    B.y = psi[base + (unsigned)(k0 + 1) * right];
                    acc = __builtin_amdgcn_wmma_f32_16x16x4_f32(
                        /*neg_a=*/false, A[kc], /*neg_b=*/false, B,
                        /*c_mod=*/(short)0, acc,
                        /*reuse_a=*/false, /*reuse_b=*/false);
                }
                // D layout: VGPR v = row v (lanes 0-15) / row v+8 (lanes 16-31)
#pragma unroll
                for (int v = 0; v < 8; ++v) {
                    psi[base + (unsigned)(v + 8 * half) * right] = acc[v];
                }
            }
            __syncthreads();
        }

        // ---- CZ ring diagonal after each variational layer ----
        if (layer >= 1) {
            for (int i = tid; i < DIM; i += THREADS) {
                const unsigned idx  = (unsigned)i;
                // adjacent index-bit pairs (qubit ring maps onto bit ring)
                const unsigned adj  = (idx & (idx >> 1)) & 0x7FFFu;
                const unsigned wrap = (idx >> 15) & idx & 1u;     // bit15 & bit0
                const unsigned par  = ((unsigned)__popc(adj) + wrap) & 1u;
                psi[i] *= 1.0f - 2.0f * (float)par;
            }
            __syncthreads();
        }
    }

    // ---- <Z_0> = sum psi^2 * (1 - 2*bit15(idx)) ----
    float accz = 0.0f;
    for (int i = tid; i < DIM; i += THREADS) {
        const float p = psi[i];
        const float z = 1.0f - 2.0f * (float)((i >> 15) & 1);
        accz += p * p * z;
    }
    red[tid] = accz;
    __syncthreads();
    for (int s = THREADS / 2; s > 0; s >>= 1) {
        if (tid < s) red[tid] += red[tid + s];
        __syncthreads();
    }
    if (tid == 0) out[m] = red[0];   // out.reshape(OUTF, NB) row-major == m
}

extern "C" void kernel_launch(void* const* d_in, const int* in_sizes, int n_in,
                              void* d_out, int out_size, void* d_ws, size_t ws_size,
                              hipStream_t stream) {
    (void)in_sizes; (void)n_in; (void)out_size; (void)d_ws; (void)ws_size;
    const float* x = (const float*)d_in[0];   // [16, 16] f32
    const float* w = (const float*)d_in[1];   // [8, 16, 4] f32
    float* out = (float*)d_out;               // [8, 16] f32

    const size_t shmem = (size_t)(DIM + 256) * sizeof(float);  // 257 KB < 320 KB/WGP
    (void)hipFuncSetAttribute((const void*)qnn_statevector_wmma,
                              hipFuncAttributeMaxDynamicSharedMemorySize,
                              (int)shmem);
    qnn_statevector_wmma<<<OUTF * NB, THREADS, shmem, stream>>>(x, w, out);
}